// AblationGNN_45973329936466
// MI455X (gfx1250) — compile-verified
//
#include <hip/hip_runtime.h>
#include <math.h>

// Problem constants (fixed by the reference)
#define NNODES 100000
#define NEDGES 600000
#define FIN    128
#define HDIM   128
#define PDIM   64
#define KDIM   128   // inner dim of every GEMM (FIN == HDIM == 128)

typedef __attribute__((ext_vector_type(2))) float v2f;
typedef __attribute__((ext_vector_type(8))) float v8f;

__device__ __forceinline__ float gelu_exact(float x) {
    // exact GELU: 0.5 x (1 + erf(x/sqrt(2)))
    return 0.5f * x * (1.0f + erff(x * 0.70710678118654752440f));
}
__device__ __forceinline__ float sigmoid_f(float x) {
    return 1.0f / (1.0f + expf(-x));
}

// ---------------------------------------------------------------------------
// utility: zero fill
// ---------------------------------------------------------------------------
__global__ void fill_zero_kernel(float* __restrict__ p, long long n) {
    long long i = (long long)blockIdx.x * blockDim.x + threadIdx.x;
    if (i < n) p[i] = 0.0f;
}

// ---------------------------------------------------------------------------
// degree / dinv
// ---------------------------------------------------------------------------
__global__ void deg_kernel(const int* __restrict__ dst, float* __restrict__ deg) {
    int e = blockIdx.x * blockDim.x + threadIdx.x;
    if (e < NEDGES) atomicAdd(&deg[dst[e]], 1.0f);
}
__global__ void dinv_kernel(float* __restrict__ deg) {
    int i = blockIdx.x * blockDim.x + threadIdx.x;
    if (i < NNODES) deg[i] = rsqrtf(deg[i] + 1.0f);   // self loop adds 1
}

// ---------------------------------------------------------------------------
// WMMA fp32 GEMM: Out[M,NC] = A[M,128] * W[128,NC]  (+ fused epilogue)
// One wave per 16-row strip computing all NC columns (NC/16 accumulators):
// each A float2 load feeds NC/16 v_wmma_f32_16x16x4_f32 ops, cutting A read
// traffic NC/16-fold vs one-tile-per-wave. W (64 KB) is WGP$/L0-resident.
// M = 100000 = 6250*16; guard is wave-uniform so EXEC is all-ones for WMMA.
// ---------------------------------------------------------------------------
enum { EP_NONE = 0, EP_BN_GELU = 1, EP_BIAS_GELU = 2, EP_BIAS = 3 };

template <int EPI, int NC>
__global__ __launch_bounds__(256) void gemm_wmma_kernel(
    const float* __restrict__ A, const float* __restrict__ W,
    float* __restrict__ Out,
    const float* __restrict__ bias,
    const float* __restrict__ bn_g, const float* __restrict__ bn_b,
    const float* __restrict__ bn_m, const float* __restrict__ bn_v) {

    const int lane = threadIdx.x & 31;
    const int wave = threadIdx.x >> 5;
    const int rt   = blockIdx.x * 8 + wave;           // row tile (16 rows)
    if (rt >= NNODES / 16) return;                    // uniform per wave

    constexpr int CT = NC / 16;                       // column tiles
    const int half = lane >> 4;                       // 0: lanes 0-15, 1: 16-31
    const int l16  = lane & 15;
    const int koff = half * 2;                        // A/B K offset per half

    const float* Arow = A + (size_t)(rt * 16 + l16) * KDIM;

    v8f acc[CT];
#pragma unroll
    for (int c = 0; c < CT; ++c) acc[c] = (v8f){};

#pragma unroll 4
    for (int k = 0; k < KDIM; k += 4) {
        // A tile 16x4: lane l16 holds row m=l16, K = koff, koff+1
        float2 av = *(const float2*)(Arow + k + koff);
        v2f a; a.x = av.x; a.y = av.y;
        const float* Wr0 = W + (size_t)(k + koff) * NC;
#pragma unroll
        for (int c = 0; c < CT; ++c) {
            const int col = c * 16 + l16;
            // B tile 4x16: lane l16 holds col, rows K = koff, koff+1
            v2f b;
            b.x = Wr0[col];
            b.y = Wr0[NC + col];
            acc[c] = __builtin_amdgcn_wmma_f32_16x16x4_f32(
                        false, a, false, b, (short)0, acc[c], false, false);
        }
    }

    // C/D layout: vgpr v -> row rt*16 + v + half*8, col = c*16 + l16
    const int rbase = rt * 16 + half * 8;
#pragma unroll
    for (int c = 0; c < CT; ++c) {
        const int col = c * 16 + l16;
#pragma unroll
        for (int v = 0; v < 8; ++v) {
            float val = acc[c][v];
            if constexpr (EPI != EP_NONE) val += bias[col];
            if constexpr (EPI == EP_BN_GELU) {
                val = (val - bn_m[col]) * rsqrtf(bn_v[col] + 1e-5f) * bn_g[col] + bn_b[col];
                val = gelu_exact(val);
            }
            if constexpr (EPI == EP_BIAS_GELU) {
                val = gelu_exact(val);
            }
            Out[(size_t)(rbase + v) * NC + col] = val;
        }
    }
}

// ---------------------------------------------------------------------------
// Aggregation init: agg = h * dinv^2 + bias  (self-loop term; replaces a
// plain zero-fill so the finalize pass never re-reads h)
// ---------------------------------------------------------------------------
__global__ void agg_init_kernel(
    const float* __restrict__ h, const float* __restrict__ dinv,
    const float* __restrict__ bias, float* __restrict__ agg) {
    long long i = (long long)blockIdx.x * blockDim.x + threadIdx.x;
    if (i >= (long long)NNODES * HDIM) return;
    int n = (int)(i >> 7);
    int f = (int)(i & 127);
    float di = dinv[n];
    agg[i] = h[i] * di * di + bias[f];
}

// ---------------------------------------------------------------------------
// Edge aggregation: agg[dst] += h[src] * dinv[src]*dinv[dst]
// One wave per edge: lanes cover the 128-float row as 32 x float4.
// h (51.2 MB) is L2-resident (192 MB L2), so gather + float atomics stay in L2.
// ---------------------------------------------------------------------------
__global__ __launch_bounds__(256) void edge_agg_kernel(
    const float* __restrict__ h, const int* __restrict__ src,
    const int* __restrict__ dst, const float* __restrict__ dinv,
    float* __restrict__ agg) {
    long long tid = (long long)blockIdx.x * blockDim.x + threadIdx.x;
    int e = (int)(tid >> 5);
    if (e >= NEDGES) return;
    int fo = ((int)tid & 31) << 2;            // feature offset 0..124
    int s = src[e], d = dst[e];
    float w = dinv[s] * dinv[d];
    float4 hv = *(const float4*)(h + (size_t)s * HDIM + fo);
    float* ap = agg + (size_t)d * HDIM + fo;
    atomicAdd(ap + 0, hv.x * w);
    atomicAdd(ap + 1, hv.y * w);
    atomicAdd(ap + 2, hv.z * w);
    atomicAdd(ap + 3, hv.w * w);
}

// ---------------------------------------------------------------------------
// GCN finalize: out = gelu(agg)   (agg already holds conv output + bias)
// ---------------------------------------------------------------------------
__global__ void gcn_finalize_kernel(
    const float* __restrict__ agg, float* __restrict__ out) {
    long long i = (long long)blockIdx.x * blockDim.x + threadIdx.x;
    if (i >= (long long)NNODES * HDIM) return;
    out[i] = gelu_exact(agg[i]);
}

// ---------------------------------------------------------------------------
// Gated fusion + classifier. One wave per node (wave32):
//   gate = sigmoid(<[h_mlp,h_graph], gate_w> + gb)
//   h_fused = gate*h_graph + (1-gate)*h_mlp      (stored)
//   logits  = <h_fused, cls_w> + cb
// ---------------------------------------------------------------------------
__global__ __launch_bounds__(256) void fuse_kernel(
    const float* __restrict__ hmlp, const float* __restrict__ hg,
    const float* __restrict__ gate_w, const float* __restrict__ gate_b,
    const float* __restrict__ cls_w, const float* __restrict__ cls_b,
    float* __restrict__ hfused, float* __restrict__ logits) {
    int node = blockIdx.x * 8 + (threadIdx.x >> 5);
    if (node >= NNODES) return;
    int lane = threadIdx.x & 31;
    int fo = lane * 4;

    const float4 m   = *(const float4*)(hmlp + (size_t)node * HDIM + fo);
    const float4 g   = *(const float4*)(hg   + (size_t)node * HDIM + fo);
    const float4 gwm = *(const float4*)(gate_w + fo);          // first H rows
    const float4 gwg = *(const float4*)(gate_w + HDIM + fo);   // second H rows

    float part = m.x * gwm.x + m.y * gwm.y + m.z * gwm.z + m.w * gwm.w
               + g.x * gwg.x + g.y * gwg.y + g.z * gwg.z + g.w * gwg.w;
#pragma unroll
    for (int off = 16; off > 0; off >>= 1) part += __shfl_xor(part, off, 32);
    float gate = sigmoid_f(part + gate_b[0]);

    float4 hf;
    hf.x = gate * g.x + (1.0f - gate) * m.x;
    hf.y = gate * g.y + (1.0f - gate) * m.y;
    hf.z = gate * g.z + (1.0f - gate) * m.z;
    hf.w = gate * g.w + (1.0f - gate) * m.w;
    *(float4*)(hfused + (size_t)node * HDIM + fo) = hf;

    const float4 cw = *(const float4*)(cls_w + fo);
    float p2 = hf.x * cw.x + hf.y * cw.y + hf.z * cw.z + hf.w * cw.w;
#pragma unroll
    for (int off = 16; off > 0; off >>= 1) p2 += __shfl_xor(p2, off, 32);
    if (lane == 0) logits[node] = p2 + cls_b[0];
}

// ---------------------------------------------------------------------------
// Launch sequence
// ---------------------------------------------------------------------------
extern "C" void kernel_launch(void* const* d_in, const int* in_sizes, int n_in,
                              void* d_out, int out_size, void* d_ws, size_t ws_size,
                              hipStream_t stream) {
    (void)in_sizes; (void)n_in; (void)out_size; (void)ws_size;

    const float* x      = (const float*)d_in[0];
    const int*   eidx   = (const int*)d_in[1];
    const int*   src    = eidx;            // edge_index[0]
    const int*   dst    = eidx + NEDGES;   // edge_index[1]
    const float* mlp_w  = (const float*)d_in[2];
    const float* mlp_b  = (const float*)d_in[3];
    const float* bn_g   = (const float*)d_in[4];
    const float* bn_b   = (const float*)d_in[5];
    const float* bn_m   = (const float*)d_in[6];
    const float* bn_v   = (const float*)d_in[7];
    const float* gcn_w[3] = { (const float*)d_in[8],  (const float*)d_in[10], (const float*)d_in[12] };
    const float* gcn_b[3] = { (const float*)d_in[9],  (const float*)d_in[11], (const float*)d_in[13] };
    const float* gate_w  = (const float*)d_in[14];
    const float* gate_b  = (const float*)d_in[15];
    const float* proj_w1 = (const float*)d_in[16];
    const float* proj_b1 = (const float*)d_in[17];
    const float* proj_w2 = (const float*)d_in[18];
    const float* proj_b2 = (const float*)d_in[19];
    const float* cls_w   = (const float*)d_in[20];
    const float* cls_b   = (const float*)d_in[21];

    // workspace layout
    const size_t nh = (size_t)NNODES * HDIM;
    float* dinv = (float*)d_ws;          // N        (holds degree first)
    float* bufM = dinv + NNODES;         // N*H  h_mlp (persists)
    float* bufH = bufM + nh;             // N*H  pre-agg h   -> later h_fused
    float* bufA = bufH + nh;             // N*H  aggregation -> later proj1 out
    float* bufG = bufA + nh;             // N*H  gcn layer output (h_graph)

    float* logits = (float*)d_out;       // [N]
    float* zproj  = logits + NNODES;     // [N,64]

    const dim3 B(256);
    const int gN    = (NNODES + 255) / 256;
    const int gE    = (NEDGES + 255) / 256;
    const int gNH   = (int)((nh + 255) / 256);                     // 50000
    const int gEdge = (int)(((long long)NEDGES * 32 + 255) / 256); // 75000
    const int gGemm = (NNODES / 16 + 7) / 8;                       // 782
    const int gFuse = NNODES / 8;                                  // 12500

    // 1) degrees -> dinv
    fill_zero_kernel<<<gN, B, 0, stream>>>(dinv, NNODES);
    deg_kernel<<<gE, B, 0, stream>>>(dst, dinv);
    dinv_kernel<<<gN, B, 0, stream>>>(dinv);

    // 2) MLP branch: gelu(BN(x @ mlp_w + b))
    gemm_wmma_kernel<EP_BN_GELU, HDIM><<<gGemm, B, 0, stream>>>(
        x, mlp_w, bufM, mlp_b, bn_g, bn_b, bn_m, bn_v);

    // 3) GCN branch: 3 x (GCNConv + GELU)
    const float* gin = x;
    for (int layer = 0; layer < 3; ++layer) {
        gemm_wmma_kernel<EP_NONE, HDIM><<<gGemm, B, 0, stream>>>(
            gin, gcn_w[layer], bufH, nullptr, nullptr, nullptr, nullptr, nullptr);
        agg_init_kernel<<<gNH, B, 0, stream>>>(bufH, dinv, gcn_b[layer], bufA);
        edge_agg_kernel<<<gEdge, B, 0, stream>>>(bufH, src, dst, dinv, bufA);
        gcn_finalize_kernel<<<gNH, B, 0, stream>>>(bufA, bufG);
        gin = bufG;
    }

    // 4) gated fusion + classifier (h_fused -> bufH, logits -> d_out[0:N])
    fuse_kernel<<<gFuse, B, 0, stream>>>(
        bufM, bufG, gate_w, gate_b, cls_w, cls_b, bufH, logits);

    // 5) projection head: z = gelu(h_fused @ W1 + b1) @ W2 + b2
    gemm_wmma_kernel<EP_BIAS_GELU, HDIM><<<gGemm, B, 0, stream>>>(
        bufH, proj_w1, bufA, proj_b1, nullptr, nullptr, nullptr, nullptr);
    gemm_wmma_kernel<EP_BIAS, PDIM><<<gGemm, B, 0, stream>>>(
        bufA, proj_w2, zproj, proj_b2, nullptr, nullptr, nullptr, nullptr);
}